// FP_21440476741674
// MI455X (gfx1250) — compile-verified
//
#include <hip/hip_runtime.h>
#include <math.h>

// Forward projection: 256^3 fp32 volume -> (256 z, 256 y, 16 views) sinogram.
// iz == z exactly, so each slice is an independent 2D bilinear problem.
// Strategy: one workgroup per z-slice; stage slice into LDS (260KB, fits the
// 320KB CDNA5 WGP LDS) with async global->LDS B128 copies; all 4.2M gathers
// per slice are then served from LDS, amortized over all 16 views.

#define VIEWS 16
#define N 256
#define LDS_PITCH 260                 // floats; 1040B rows: 16B-aligned, pitch%64==4 banks
#define LDS_BYTES (N * LDS_PITCH * 4) // 266240 B < 320KB WGP LDS

__global__ __launch_bounds__(1024)
void fp_project_kernel(const float* __restrict__ vol, float* __restrict__ out) {
    const int z   = blockIdx.x;
    const int tid = threadIdx.x;

    extern __shared__ float lds[];
    // Low 32 bits of the generic (shared-aperture) address == LDS byte offset.
    const unsigned ldsBase = (unsigned)(unsigned long long)(size_t)lds;

    const float* __restrict__ slice = vol + (size_t)z * (N * N);

    // ---- Stage slice into LDS via async B128 copies (ASYNCcnt-tracked). ----
    // 65536 floats = 16384 16B chunks; 1024 threads x 16 iterations.
#pragma unroll
    for (int i = 0; i < 16; ++i) {
        const int idx4 = i * 1024 + tid;        // 16B chunk id, row-major
        const int r    = idx4 >> 6;             // row 0..255
        const int c4   = idx4 & 63;             // 16B chunk within row
        const unsigned lofs = ldsBase + (unsigned)r * (LDS_PITCH * 4u)
                                      + ((unsigned)c4 << 4);
        const float* g = slice + ((size_t)idx4 << 2);
        asm volatile("global_load_async_to_lds_b128 %0, %1, off"
                     :
                     : "v"(lofs), "v"(g)
                     : "memory");
    }
    asm volatile("s_wait_asynccnt 0x0" ::: "memory");
    __syncthreads();

    // ---- Compute: each thread owns 4 (view,row) outputs of this slice. ----
    const float inv256 = 1.0f / 256.0f;
    const float unit   = 3.14159265358979323846f / 16.0f;  // pi/VIEWS
    const float a0     = 3.14159265358979323846f / 180.0f; // START_VIEW * pi/180

#pragma unroll
    for (int chunk = 0; chunk < 4; ++chunk) {
        const int o = chunk * 1024 + tid;   // 0..4095
        const int v = o >> 8;               // view 0..15
        const int y = o & 255;              // output row

        const float ang = a0 + (float)v * unit;
        float s, c;
        sincosf(ang, &s, &c);

        const float gy = (2.0f * (float)y + 1.0f) * inv256 - 1.0f;
        // ix = (c*gx - s*gy + 1)*128 - 0.5 ; iy = (s*gx + c*gy + 1)*128 - 0.5
        const float bix = (-s * gy + 1.0f) * 128.0f - 0.5f;
        const float biy = ( c * gy + 1.0f) * 128.0f - 0.5f;

        float acc = 0.0f;
#pragma unroll 4
        for (int x = 0; x < 256; ++x) {
            const float gx = (2.0f * (float)x + 1.0f) * inv256 - 1.0f;
            const float ix = fmaf(c * gx, 128.0f, bix);
            const float iy = fmaf(s * gx, 128.0f, biy);
            // Outside the slice entirely (zero-padding) -> contributes nothing.
            if (ix <= -1.0f || ix >= 256.0f || iy <= -1.0f || iy >= 256.0f)
                continue;

            const float x0f = floorf(ix);
            const float y0f = floorf(iy);
            const float wx  = ix - x0f;
            const float wy  = iy - y0f;
            const int x0 = (int)x0f, y0 = (int)y0f;
            const int x1 = x0 + 1,   y1 = y0 + 1;

            const bool vx0 = (x0 >= 0) && (x0 < N);
            const bool vx1 = (x1 >= 0) && (x1 < N);
            const bool vy0 = (y0 >= 0) && (y0 < N);
            const bool vy1 = (y1 >= 0) && (y1 < N);

            const int cx0 = min(max(x0, 0), N - 1);
            const int cx1 = min(max(x1, 0), N - 1);
            const int cy0 = min(max(y0, 0), N - 1);
            const int cy1 = min(max(y1, 0), N - 1);

            const float v00 = lds[cy0 * LDS_PITCH + cx0];
            const float v01 = lds[cy0 * LDS_PITCH + cx1];
            const float v10 = lds[cy1 * LDS_PITCH + cx0];
            const float v11 = lds[cy1 * LDS_PITCH + cx1];

            const float w00 = (vx0 && vy0) ? (1.0f - wx) * (1.0f - wy) : 0.0f;
            const float w01 = (vx1 && vy0) ? wx * (1.0f - wy)          : 0.0f;
            const float w10 = (vx0 && vy1) ? (1.0f - wx) * wy          : 0.0f;
            const float w11 = (vx1 && vy1) ? wx * wy                   : 0.0f;

            acc = fmaf(w00, v00, acc);
            acc = fmaf(w01, v01, acc);
            acc = fmaf(w10, v10, acc);
            acc = fmaf(w11, v11, acc);
        }
        out[((size_t)z * N + y) * VIEWS + v] = acc;
    }
}

extern "C" void kernel_launch(void* const* d_in, const int* in_sizes, int n_in,
                              void* d_out, int out_size, void* d_ws, size_t ws_size,
                              hipStream_t stream) {
    (void)in_sizes; (void)n_in; (void)d_ws; (void)ws_size; (void)out_size;
    const float* x = (const float*)d_in[0];
    float* out = (float*)d_out;
    fp_project_kernel<<<dim3(N), dim3(1024), LDS_BYTES, stream>>>(x, out);
}